// NormalLoss_8117488189450
// MI455X (gfx1250) — compile-verified
//
#include <hip/hip_runtime.h>
#include <hip/hip_bf16.h>
#include <math.h>

#define NPTS   4096
#define NBATCH 8
#define KNN    10
#define QPW    16          // queries per wave (one WMMA N-tile)
#define WAVES  8           // waves per workgroup
#define QPB    (QPW*WAVES) // 128 queries per workgroup
#define NELEM  (NBATCH*NPTS*3)
#define RBLK   96          // reduction blocks

typedef float v2f __attribute__((ext_vector_type(2)));
typedef float v8f __attribute__((ext_vector_type(8)));

// Branch-free sorted insert of a packed (dist|idx) key into an ascending list.
// Each step lowers to v_cmp_lt_u32 + 2x v_cndmask -- no EXEC manipulation.
__device__ __forceinline__ void topk_insert_key(unsigned (&keys)[KNN], unsigned key) {
    if (key < keys[KNN - 1]) {
#pragma unroll
        for (int j = 0; j < KNN; ++j) {
            const unsigned tj = keys[j];
            const bool sm = key < tj;
            keys[j] = sm ? key : tj;
            key     = sm ? tj  : key;
        }
    }
}

// One workgroup: one batch slice of 128 query points vs all 4096 candidates.
// grid = (NPTS/QPB, NBATCH), block = 256 (8 waves).
__global__ __launch_bounds__(256)
void normals_kernel(const float* __restrict__ x, float* __restrict__ out_n) {
    // x layout: (B, 3, N).  LDS: padded points (x,y,z,|p|^2), 16B stride
    // (bank-conflict-free for the 16-lane A-tile ds_load_b64).
    __shared__ float pts[NPTS * 4];                 // 64 KB
    const int b   = blockIdx.y;
    const int tid = threadIdx.x;
    const float* xb = x + (size_t)b * 3 * NPTS;

    for (int n = tid; n < NPTS; n += blockDim.x) {
        float px = xb[n], py = xb[NPTS + n], pz = xb[2 * NPTS + n];
        pts[4 * n + 0] = px; pts[4 * n + 1] = py; pts[4 * n + 2] = pz;
        pts[4 * n + 3] = px * px + py * py + pz * pz;
    }
    __syncthreads();

    const int wave  = tid >> 5;
    const int lane  = tid & 31;
    const int lhalf = lane >> 4;       // 0: K=0,1 / rows M=v ; 1: K=2,3 / rows M=v+8
    const int l16   = lane & 15;
    const int q     = blockIdx.x * QPB + wave * QPW + l16;   // this lane's query (col N)

    // B operand (4x16, col = query): rows are (-2x, -2y, -2z, 1).
    v2f bmat;
    const float sqq = pts[4 * q + 3];
    bmat[0] = -2.0f * pts[4 * q + 2 * lhalf];
    bmat[1] = lhalf ? 1.0f : (-2.0f * pts[4 * q + 1]);

    // C accumulator = |p_q|^2 broadcast (adds the query norm to every row).
    v8f cacc;
#pragma unroll
    for (int v = 0; v < 8; ++v) cacc[v] = sqq;

    unsigned keys[KNN];
#pragma unroll
    for (int j = 0; j < KNN; ++j) keys[j] = 0xFFFFFFFFu;

    // Software-pipelined sweep over candidate 16-row tiles:
    // prefetch A(t+1) from LDS, WMMA(t), then 8 branch-free inserts.
    v2f a_cur;
    a_cur[0] = pts[4 * l16 + 2 * lhalf];
    a_cur[1] = pts[4 * l16 + 2 * lhalf + 1];
#pragma unroll 2
    for (int t = 0; t < NPTS / 16; ++t) {
        const int mn = (((t + 1) & (NPTS / 16 - 1)) * 16) + l16;
        v2f a_next;
        a_next[0] = pts[4 * mn + 2 * lhalf];
        a_next[1] = pts[4 * mn + 2 * lhalf + 1];
        const v8f d = __builtin_amdgcn_wmma_f32_16x16x4_f32(
            false, a_cur, false, bmat, (short)0, cacc, false, false);
#pragma unroll
        for (int v = 0; v < 8; ++v) {
            // clamp: self-distance may round to -eps; keep unsigned-monotone.
            const float dv = fmaxf(d[v], 0.0f);
            const unsigned key = (__float_as_uint(dv) & 0xFFFFF000u) |
                                 (unsigned)(t * 16 + v + 8 * lhalf);
            topk_insert_key(keys, key);
        }
        a_cur = a_next;
    }

    // Merge the two half-lists for this query (lanes L and L^16).
    unsigned ok[KNN];
#pragma unroll
    for (int j = 0; j < KNN; ++j)
        ok[j] = (unsigned)__shfl_xor((int)keys[j], 16, 32);
#pragma unroll
    for (int j = 0; j < KNN; ++j) topk_insert_key(keys, ok[j]);

    // Gather the 10 neighbors, covariance, smallest eigenvector (analytic 3x3).
    float nx[KNN], ny[KNN], nz[KNN];
    float mx = 0.f, my = 0.f, mz = 0.f;
#pragma unroll
    for (int j = 0; j < KNN; ++j) {
        const int id = (int)(keys[j] & 0xFFFu);
        nx[j] = pts[4 * id + 0]; ny[j] = pts[4 * id + 1]; nz[j] = pts[4 * id + 2];
        mx += nx[j]; my += ny[j]; mz += nz[j];
    }
    const float invk = 1.0f / (float)KNN;
    mx *= invk; my *= invk; mz *= invk;
    float cxx = 0.f, cxy = 0.f, cxz = 0.f, cyy = 0.f, cyz = 0.f, czz = 0.f;
#pragma unroll
    for (int j = 0; j < KNN; ++j) {
        const float dx = nx[j] - mx, dy = ny[j] - my, dz = nz[j] - mz;
        cxx += dx * dx; cxy += dx * dy; cxz += dx * dz;
        cyy += dy * dy; cyz += dy * dz; czz += dz * dz;
    }
    cxx *= invk; cxy *= invk; cxz *= invk; cyy *= invk; cyz *= invk; czz *= invk;

    // Smallest eigenvalue via trigonometric closed form.
    const float qm = (cxx + cyy + czz) * (1.0f / 3.0f);
    const float p1 = cxy * cxy + cxz * cxz + cyz * cyz;
    const float p2 = (cxx - qm) * (cxx - qm) + (cyy - qm) * (cyy - qm) +
                     (czz - qm) * (czz - qm) + 2.0f * p1;
    const float p = sqrtf(p2 * (1.0f / 6.0f));
    float eig = qm;
    if (p > 1e-12f) {
        const float ip  = 1.0f / p;
        const float bxx = (cxx - qm) * ip, byy = (cyy - qm) * ip, bzz = (czz - qm) * ip;
        const float bxy = cxy * ip, bxz = cxz * ip, byz = cyz * ip;
        float r = 0.5f * (bxx * (byy * bzz - byz * byz)
                        - bxy * (bxy * bzz - byz * bxz)
                        + bxz * (bxy * byz - byy * bxz));
        r = fminf(1.0f, fmaxf(-1.0f, r));
        const float phi = acosf(r) * (1.0f / 3.0f);
        eig = qm + 2.0f * p * cosf(phi + 2.0943951023931953f);  // + 2*pi/3 -> smallest
    }
    // Eigenvector: best cross product of rows of (C - eig*I).
    const float r0x = cxx - eig, r0y = cxy,       r0z = cxz;
    const float r1x = cxy,       r1y = cyy - eig, r1z = cyz;
    const float r2x = cxz,       r2y = cyz,       r2z = czz - eig;
    float ax = r0y * r1z - r0z * r1y, ay = r0z * r1x - r0x * r1z, az = r0x * r1y - r0y * r1x;
    float bx = r0y * r2z - r0z * r2y, by = r0z * r2x - r0x * r2z, bz = r0x * r2y - r0y * r2x;
    float gx = r1y * r2z - r1z * r2y, gy = r1z * r2x - r1x * r2z, gz = r1x * r2y - r1y * r2x;
    const float na = ax * ax + ay * ay + az * az;
    const float nb = bx * bx + by * by + bz * bz;
    const float ng = gx * gx + gy * gy + gz * gz;
    float vx = ax, vy = ay, vz = az, nn = na;
    if (nb > nn) { vx = bx; vy = by; vz = bz; nn = nb; }
    if (ng > nn) { vx = gx; vy = gy; vz = gz; nn = ng; }
    if (nn < 1e-30f) { vx = 1.0f; vy = 0.0f; vz = 0.0f; nn = 1.0f; }
    const float inorm = rsqrtf(nn);
    vx *= inorm; vy *= inorm; vz *= inorm;
    // Deterministic sign: largest-|component| positive.
    const float axv = fabsf(vx), ayv = fabsf(vy), azv = fabsf(vz);
    float key2 = vx;
    if (ayv > axv && ayv >= azv) key2 = vy;
    else if (azv > axv && azv > ayv) key2 = vz;
    if (key2 < 0.0f) { vx = -vx; vy = -vy; vz = -vz; }

    if (lhalf == 0) {   // lanes 0-15 store (lanes 16-31 computed identical merge)
        float* o = out_n + ((size_t)b * NPTS + q) * 3;
        o[0] = vx; o[1] = vy; o[2] = vz;
    }
}

__global__ __launch_bounds__(256)
void loss_partial_kernel(const float* __restrict__ a, const float* __restrict__ b,
                         float* __restrict__ part) {
    __shared__ float red[256];
    float s = 0.0f;
    for (int i = blockIdx.x * blockDim.x + threadIdx.x; i < NELEM;
         i += gridDim.x * blockDim.x) {
        const float d = a[i] - b[i];
        s += d * d;
    }
    red[threadIdx.x] = s;
    __syncthreads();
    for (int off = 128; off > 0; off >>= 1) {
        if (threadIdx.x < off) red[threadIdx.x] += red[threadIdx.x + off];
        __syncthreads();
    }
    if (threadIdx.x == 0) part[blockIdx.x] = red[0];
}

__global__ void loss_final_kernel(const float* __restrict__ part, float* __restrict__ out) {
    float s = 0.0f;
    for (int i = 0; i < RBLK; ++i) s += part[i];   // fixed order -> deterministic
    out[0] = s * (1.0f / (float)NELEM);
}

extern "C" void kernel_launch(void* const* d_in, const int* in_sizes, int n_in,
                              void* d_out, int out_size, void* d_ws, size_t ws_size,
                              hipStream_t stream) {
    const float* pred = (const float*)d_in[0];
    const float* gt   = (const float*)d_in[1];
    float* np   = (float*)d_ws;            // pred normals: 98304 floats
    float* ngt  = np + NELEM;              // gt   normals: 98304 floats
    float* part = ngt + NELEM;             // RBLK partial sums

    dim3 grid(NPTS / QPB, NBATCH);
    normals_kernel<<<grid, 256, 0, stream>>>(pred, np);
    normals_kernel<<<grid, 256, 0, stream>>>(gt,   ngt);
    loss_partial_kernel<<<RBLK, 256, 0, stream>>>(np, ngt, part);
    loss_final_kernel<<<1, 1, 0, stream>>>(part, (float*)d_out);
}